// BoxIoU_4363686773475
// MI455X (gfx1250) — compile-verified
//
#include <hip/hip_runtime.h>
#include <hip/hip_bf16.h>

// ---- native vector types (real clang vectors so __builtin_nontemporal_load works) ----
typedef __attribute__((ext_vector_type(4))) float v4f;
typedef __attribute__((ext_vector_type(2))) float v2f;
typedef __attribute__((ext_vector_type(8))) float v8f;

// Wave32 all-lane reduction via V_WMMA_F32_16X16X4_F32.
// A layout (32-bit A, 16x4): VGPR0 = {lanes0-15: K=0, lanes16-31: K=2},
//                            VGPR1 = {lanes0-15: K=1, lanes16-31: K=3}.
// We set a = {p, 0} -> lane L<16 places p at A[L][0], lane L>=16 at A[L-16][2].
// B = all-ones 4x16 (layout-independent since every element is 1).
// D[m][n] = p_m + p_{m+16}. Summing the 8 D-VGPRs in a lane yields the sum of
// lanes {0..7,16..23} (low half-columns) or {8..15,24..31} (high), and one
// shfl_xor(16) produces the full 32-lane total in every lane.
__device__ __forceinline__ float wave_reduce_wmma(float p) {
    v2f a; a.x = p;    a.y = 0.0f;
    v2f b; b.x = 1.0f; b.y = 1.0f;
    v8f c = {};
    c = __builtin_amdgcn_wmma_f32_16x16x4_f32(
            /*neg_a=*/false, a, /*neg_b=*/false, b,
            /*c_mod=*/(short)0, c, /*reuse_a=*/false, /*reuse_b=*/false);
    float s = ((c[0] + c[1]) + (c[2] + c[3])) + ((c[4] + c[5]) + (c[6] + c[7]));
    s += __shfl_xor(s, 16, 32);   // combine the two half-wave sums
    return s;                     // total across all 32 lanes, in every lane
}

// Phase 1: stream boxes (128-bit NT loads), per-lane IoU accumulation,
// WMMA wave reduce, fixed-order LDS block reduce, one partial per block.
__global__ __launch_bounds__(256) void iou_partial_kernel(
        const v4f* __restrict__ bp, const v4f* __restrict__ bt,
        float* __restrict__ partials, int n) {
    __shared__ float lds[8];
    const int tid    = threadIdx.x;
    const int stride = gridDim.x * blockDim.x;
    float acc = 0.0f;
    for (int i = blockIdx.x * blockDim.x + tid; i < n; i += stride) {
        const v4f p = __builtin_nontemporal_load(&bp[i]);   // x1 y1 x2 y2
        const v4f t = __builtin_nontemporal_load(&bt[i]);
        const float ap = (p.z - p.x) * (p.w - p.y);
        const float at = (t.z - t.x) * (t.w - t.y);
        const float lx = fmaxf(p.x, t.x);
        const float ly = fmaxf(p.y, t.y);
        const float rx = fminf(p.z, t.z);
        const float ry = fminf(p.w, t.w);
        const float w  = fmaxf(rx - lx, 0.0f);
        const float h  = fmaxf(ry - ly, 0.0f);
        const float inter = w * h;
        const float uni   = ap + at - inter;
        acc += inter / uni;
    }
    const float wsum = wave_reduce_wmma(acc);     // EXEC all-ones here (reconverged)
    if ((tid & 31) == 0) lds[tid >> 5] = wsum;
    __syncthreads();
    if (tid == 0) {
        float s = 0.0f;
        const int nw = blockDim.x >> 5;
        for (int w = 0; w < nw; ++w) s += lds[w]; // fixed order -> deterministic
        partials[blockIdx.x] = s;
    }
}

// Phase 2: single block reduces the block partials in a fixed order and
// writes mean = sum * (1/N). No atomics -> bitwise deterministic.
__global__ __launch_bounds__(256) void iou_final_kernel(
        const float* __restrict__ partials, float* __restrict__ out,
        int num_partials, float inv_n) {
    __shared__ float lds[8];
    const int tid = threadIdx.x;
    float acc = 0.0f;
    for (int i = tid; i < num_partials; i += blockDim.x) acc += partials[i];
    const float wsum = wave_reduce_wmma(acc);
    if ((tid & 31) == 0) lds[tid >> 5] = wsum;
    __syncthreads();
    if (tid == 0) {
        float s = 0.0f;
        for (int w = 0; w < 8; ++w) s += lds[w];
        out[0] = s * inv_n;
    }
}

extern "C" void kernel_launch(void* const* d_in, const int* in_sizes, int n_in,
                              void* d_out, int out_size, void* d_ws, size_t ws_size,
                              hipStream_t stream) {
    // inputs: [0] masks_pred(8) unused, [1] masks_true(8) unused,
    //         [2] boxes_pred (N*4 f32), [3] boxes_true (N*4 f32)
    const int n = in_sizes[2] / 4;                 // number of box pairs
    const v4f* bp = (const v4f*)d_in[2];
    const v4f* bt = (const v4f*)d_in[3];
    float* partials = (float*)d_ws;
    float* out = (float*)d_out;

    const int block = 256;
    int blocks = 2048;                             // ~16k waves: saturates 23.3 TB/s HBM
    const int needed = (n + block - 1) / block;
    if (blocks > needed) blocks = needed;
    const int ws_cap = (int)(ws_size / sizeof(float));
    if (blocks > ws_cap) blocks = ws_cap;
    if (blocks < 1) blocks = 1;

    iou_partial_kernel<<<blocks, block, 0, stream>>>(bp, bt, partials, n);
    iou_final_kernel<<<1, 256, 0, stream>>>(partials, out, blocks,
                                            1.0f / (float)n);
}